// SlidingWindowHead_58849641890633
// MI455X (gfx1250) — compile-verified
//
#include <hip/hip_runtime.h>
#include <math.h>

typedef __attribute__((ext_vector_type(2))) float v2f;
typedef __attribute__((ext_vector_type(8))) float v8f;

#define T_SEQ   4096
#define DMODEL  512
#define HDIM    64
#define NB      2
#define BAND    272      // 17 key tiles of 16
#define SSTRIDE 273      // pad: 273 mod 64 = 17 (odd) -> conflict-free column reads

// ---------------------------------------------------------------------------
// Kernel 1: QKV projection.  One wave computes one 16x16 tile of one of Q/K/V.
// grid = (BT/16, H/16, 3), block = 32 (one wave32).
// Uses exact-fp32 matrix path: v_wmma_f32_16x16x4_f32, K swept 512/4 = 128x.
// ---------------------------------------------------------------------------
__global__ __launch_bounds__(32) void qkv_proj_kernel(
    const float* __restrict__ X,  const float* __restrict__ Wq,
    const float* __restrict__ Wk, const float* __restrict__ Wv,
    float* __restrict__ Q, float* __restrict__ K, float* __restrict__ V)
{
  const int lane = threadIdx.x & 31;
  const int l16  = lane & 15;
  const int half = lane >> 4;          // 0: K={0,1} of chunk, 1: K={2,3}
  const int mt   = blockIdx.x;         // row tile over B*T
  const int nt   = blockIdx.y;         // 16-col tile of H
  const int w    = blockIdx.z;         // 0=Q,1=K,2=V

  const float* W = (w == 0) ? Wq : (w == 1 ? Wk : Wv);
  float*       O = (w == 0) ? Q  : (w == 1 ? K  : V);

  const float* Xrow = X + (size_t)(mt * 16 + l16) * DMODEL;    // A row = l16
  const float* Wcol = W + nt * 16 + l16;                       // B col = l16

  v8f acc = {};
  #pragma unroll 4
  for (int k = 0; k < DMODEL; k += 4) {
    const int ka = k + half * 2;
    v2f a, b;
    a.x = Xrow[ka];
    a.y = Xrow[ka + 1];
    b.x = Wcol[(size_t)ka * HDIM];
    b.y = Wcol[(size_t)(ka + 1) * HDIM];
    acc = __builtin_amdgcn_wmma_f32_16x16x4_f32(
        false, a, false, b, (short)0, acc, false, false);
  }
  // C/D layout: VGPR r -> (M=r, N=lane) lanes 0-15 ; (M=r+8, N=lane-16) lanes 16-31
  #pragma unroll
  for (int r = 0; r < 8; ++r) {
    const int M = r + half * 8;
    O[(size_t)(mt * 16 + M) * HDIM + nt * 16 + l16] = acc[r];
  }
}

// ---------------------------------------------------------------------------
// Kernel 2: banded sliding-window attention.  One wave per 16-row query tile.
// Scores over the 272-wide key band -> masked softmax in LDS -> att x V.
// grid = BT/16, block = 32.
// ---------------------------------------------------------------------------
__global__ __launch_bounds__(32) void swa_kernel(
    const float* __restrict__ Q, const float* __restrict__ K,
    const float* __restrict__ V, float* __restrict__ Out)
{
  __shared__ float S[16][SSTRIDE];

  const int lane = threadIdx.x & 31;
  const int l16  = lane & 15;
  const int half = lane >> 4;
  const int mt   = blockIdx.x;                 // tile over B*T
  const int b    = mt / (T_SEQ / 16);
  const int t0   = (mt % (T_SEQ / 16)) * 16;   // batch-local first query row
  const size_t base = (size_t)b * T_SEQ * HDIM;
  const float* Qb = Q + base;
  const float* Kb = K + base;
  const float* Vb = V + base;
  const float NEG_INF = -__builtin_inff();

  // Preload Q fragments: qa[kc] = features {4kc+2*half, +1} of row t0+l16.
  v2f qa[16];
  {
    const float* qrow = Qb + (size_t)(t0 + l16) * HDIM + half * 2;
    #pragma unroll
    for (int kc = 0; kc < 16; ++kc) {
      qa[kc].x = qrow[kc * 4];
      qa[kc].y = qrow[kc * 4 + 1];
    }
  }

  // ---- Phase 1: score band S[M][c], c = key_loc - (t0-256) in [0,272) ----
  for (int u = 0; u < 17; ++u) {
    const int ktloc = t0 - 256 + u * 16;       // batch-local key tile start
    const int keyn  = ktloc + l16;             // this lane's B-matrix column key
    const bool kin  = (keyn >= 0);             // reference zero-pads key<0
    const float* krow = Kb + (size_t)(kin ? keyn : 0) * HDIM + half * 2;

    v8f acc = {};
    #pragma unroll
    for (int kc = 0; kc < 16; ++kc) {
      v2f bf;
      bf.x = kin ? krow[kc * 4]     : 0.0f;
      bf.y = kin ? krow[kc * 4 + 1] : 0.0f;
      acc = __builtin_amdgcn_wmma_f32_16x16x4_f32(
          false, qa[kc], false, bf, (short)0, acc, false, false);
    }
    #pragma unroll
    for (int r = 0; r < 8; ++r) {
      const int M   = r + half * 8;
      const int tq  = t0 + M;
      const int key = ktloc + l16;
      // reference: j in [0,256) AND j <= t  <=>  t-255 <= key <= min(t, 2t-255)
      const bool valid = (key >= tq - 255) && (key <= tq) && (key <= 2 * tq - 255);
      S[M][u * 16 + l16] = valid ? acc[r] * 0.125f : NEG_INF;   // 1/sqrt(64)
    }
  }
  __syncthreads();

  // ---- Phase 2: softmax across band; each half-lane owns 136 columns ----
  {
    const int c0 = half * 136;
    float m = NEG_INF;
    for (int c = 0; c < 136; ++c) m = fmaxf(m, S[l16][c0 + c]);
    m = fmaxf(m, __shfl_xor(m, 16, 32));
    float sum = 0.0f;
    for (int c = 0; c < 136; ++c) {
      const float e = __expf(S[l16][c0 + c] - m);   // exp(-inf)=0 on masked
      S[l16][c0 + c] = e;
      sum += e;
    }
    sum += __shfl_xor(sum, 16, 32);
    const float inv = 1.0f / sum;
    for (int c = 0; c < 136; ++c) S[l16][c0 + c] *= inv;
  }
  __syncthreads();

  // ---- Phase 3: out(16x64) = att(16x272) x Vwin(272x64) ----
  const int kb0 = t0 - 256;
  #pragma unroll
  for (int nt = 0; nt < 4; ++nt) {
    v8f acc = {};
    for (int c4 = 0; c4 < 68; ++c4) {
      const int kr = c4 * 4 + half * 2;        // band row pair for this lane
      v2f a, bf;
      a.x = S[l16][kr];
      a.y = S[l16][kr + 1];
      const int k0 = kb0 + kr;
      bf.x = (k0     >= 0) ? Vb[(size_t)k0       * HDIM + nt * 16 + l16] : 0.0f;
      bf.y = (k0 + 1 >= 0) ? Vb[(size_t)(k0 + 1) * HDIM + nt * 16 + l16] : 0.0f;
      acc = __builtin_amdgcn_wmma_f32_16x16x4_f32(
          false, a, false, bf, (short)0, acc, false, false);
    }
    #pragma unroll
    for (int r = 0; r < 8; ++r) {
      const int M = r + half * 8;
      Out[base + (size_t)(t0 + M) * HDIM + nt * 16 + l16] = acc[r];
    }
  }
}

// ---------------------------------------------------------------------------
extern "C" void kernel_launch(void* const* d_in, const int* in_sizes, int n_in,
                              void* d_out, int out_size, void* d_ws, size_t ws_size,
                              hipStream_t stream) {
  (void)in_sizes; (void)n_in; (void)out_size; (void)ws_size;
  const float* X  = (const float*)d_in[0];
  const float* Wq = (const float*)d_in[1];
  const float* Wk = (const float*)d_in[2];
  const float* Wv = (const float*)d_in[3];
  float* out = (float*)d_out;
  float* ws  = (float*)d_ws;

  const int BT = NB * T_SEQ;                   // 8192 rows
  float* Qw = ws;
  float* Kw = ws + (size_t)BT * HDIM;
  float* Vw = ws + 2 * (size_t)BT * HDIM;      // total 6 MB of d_ws

  dim3 g1(BT / 16, HDIM / 16, 3);
  qkv_proj_kernel<<<g1, 32, 0, stream>>>(X, Wq, Wk, Wv, Qw, Kw, Vw);
  swa_kernel<<<BT / 16, 32, 0, stream>>>(Qw, Kw, Vw, out);
}